// Add_PairLinears_27504970564317
// MI455X (gfx1250) — compile-verified
//
#include <hip/hip_runtime.h>
#include <cstdint>
#include <cstddef>

// Problem constants (from reference)
#define B_    8192
#define D_    4096
#define A_    8
#define NB_   (D_/2)          // 2048 pairs
#define ROWS  8               // batch rows staged per workgroup (128 KB LDS)
#define TPB   256             // 8 waves of 32
#define NPT   2               // adjacent pairs per thread per iteration (float4 I/O)
#define ITERS (NB_/(TPB*NPT)) // 4 iterations over pair-columns

#define USE_PK_ASM 1          // emit v_pk_fma_f32 (VOP3P op 31, CDNA5 ISA §15.10)

#if defined(__has_builtin)
#  if __has_builtin(__builtin_amdgcn_global_load_async_to_lds_b32)
#    define HAVE_ASYNC_COPY 1
#  endif
#endif
#ifndef HAVE_ASYNC_COPY
#  define HAVE_ASYNC_COPY 0
#endif

typedef int   v4i_t __attribute__((ext_vector_type(4)));
typedef float v4f_t __attribute__((ext_vector_type(4)));
typedef float v2f_t __attribute__((ext_vector_type(2)));
#define AS1 __attribute__((address_space(1)))
#define AS3 __attribute__((address_space(3)))

// d = a * broadcast(w.lo) + c   (both result halves read src1's LOW half)
static __device__ __forceinline__ v2f_t pk_fma_b0(v2f_t a, v2f_t w, v2f_t c) {
#if USE_PK_ASM
    v2f_t d;
    asm("v_pk_fma_f32 %0, %1, %2, %3 op_sel:[0,0,0] op_sel_hi:[1,0,1]"
        : "=v"(d) : "v"(a), "v"(w), "v"(c));
    return d;
#else
    return (v2f_t){fmaf(a.x, w.x, c.x), fmaf(a.y, w.x, c.y)};
#endif
}
// d = a * broadcast(w.hi) + c   (both result halves read src1's HIGH half)
static __device__ __forceinline__ v2f_t pk_fma_b1(v2f_t a, v2f_t w, v2f_t c) {
#if USE_PK_ASM
    v2f_t d;
    asm("v_pk_fma_f32 %0, %1, %2, %3 op_sel:[0,1,0] op_sel_hi:[1,1,1]"
        : "=v"(d) : "v"(a), "v"(w), "v"(c));
    return d;
#else
    return (v2f_t){fmaf(a.x, w.y, c.x), fmaf(a.y, w.y, c.y)};
#endif
}

#define LO2(v) __builtin_shufflevector(v, v, 0, 1)   // aligned low pair of a quad
#define HI2(v) __builtin_shufflevector(v, v, 2, 3)   // aligned high pair
// row-pair g of the 8 staged rows: 0,1 from the low quad, 2,3 from the high quad
#define GPART(lo4, hi4, g) \
    ((g) == 0 ? LO2(lo4) : (g) == 1 ? HI2(lo4) : (g) == 2 ? LO2(hi4) : HI2(hi4))

// WG = 8 batch rows. LDS layout is 4-row interleaved: v4f chunk xs4[(r/4)*D + c]
// holds rows (4*(r/4)..+3) of column c. Async b32 DMA does the interleave during
// staging (per-lane LDS addresses); gathers are ds_load_b128 (2 per column for
// all 8 rows); FMAs are v_pk_fma_f32 with op_sel scalar broadcast (no splat movs).
__global__ __launch_bounds__(TPB) void pairlin_kernel(const float* __restrict__ x,
                                                      const float* __restrict__ W,
                                                      const int*   __restrict__ perms,
                                                      const float* __restrict__ bsum,
                                                      float* __restrict__ y) {
    __shared__ float xs[ROWS * D_];          // 128 KB (gfx1250: 320 KB/WGP)
    const int tid  = threadIdx.x;
    const int row0 = blockIdx.x * ROWS;

    // ---- Stage x[row0 .. row0+ROWS) into interleaved LDS ----
    const float* src = x + (size_t)row0 * D_;
    constexpr int CH32 = (ROWS * D_) / TPB;         // 128 b32 ops per thread
#pragma unroll
    for (int k = 0; k < CH32; ++k) {
        const int r   = k >> 4;                     // compile-time per k
        const int c   = ((k & 15) << 8) + tid;
        const int e   = k * TPB + tid;              // = r*D_ + c
        const int off = (((r >> 2) * D_ + c) << 2) + (r & 3);
#if HAVE_ASYNC_COPY
        __builtin_amdgcn_global_load_async_to_lds_b32(
            (AS1 int*)(int*)(src + e),
            (AS3 int*)(int*)(xs + off),
            0, 0);
#else
        xs[off] = src[e];
#endif
    }
#if HAVE_ASYNC_COPY
#  if defined(__has_builtin) && __has_builtin(__builtin_amdgcn_s_wait_asynccnt)
    __builtin_amdgcn_s_wait_asynccnt(0);
#  else
    asm volatile("s_wait_asynccnt 0" ::: "memory");
#  endif
#endif
    __syncthreads();

    const v4f_t* xs4 = (const v4f_t*)xs;            // xs4[idx]: rows 0-3 of col idx
                                                    // xs4[D_+idx]: rows 4-7
    // ---- Compute: 4 iterations x 2 adjacent pairs per thread ----
    for (int p = 0; p < ITERS; ++p) {
        const int pair0 = p * (TPB * NPT) + 2 * tid;     // first of 2 pairs
        const int col0  = 2 * pair0;                     // 4*tid + p*1024

        // Prefetch next iteration's tables (small, L2-resident; warm WGP$).
        if (p + 1 < ITERS) {
            const int np = pair0 + TPB * NPT;
#pragma unroll
            for (int a = 0; a < A_; ++a) {
                __builtin_prefetch(perms + (size_t)a * D_ + 2 * np, 0, 0);
                __builtin_prefetch(W + (((size_t)a * NB_ + np) << 2), 0, 0);
            }
        }

        // acc[c][g]: output column col0+c, row pair g (rows 2g, 2g+1)
        v2f_t acc[4][4];
#pragma unroll
        for (int c = 0; c < 4; ++c)
#pragma unroll
            for (int g = 0; g < 4; ++g) acc[c][g] = (v2f_t){0.f, 0.f};

#pragma unroll
        for (int a = 0; a < A_; ++a) {
            const v4i_t idx = *(const v4i_t*)(perms + (size_t)a * D_ + col0);
            const float* wp = W + (((size_t)a * NB_ + pair0) << 2);
            const v4f_t w0 = *(const v4f_t*)(wp);      // pair0: i0o0,i0o1,i1o0,i1o1
            const v4f_t w1 = *(const v4f_t*)(wp + 4);  // pair1
            const v2f_t w0L = LO2(w0), w0H = HI2(w0);  // aligned pairs, free
            const v2f_t w1L = LO2(w1), w1H = HI2(w1);

            const v4f_t g0  = xs4[idx.x];              // rows 0-3 of input col idx.x
            const v4f_t g0h = xs4[D_ + idx.x];         // rows 4-7
            const v4f_t g1  = xs4[idx.y];
            const v4f_t g1h = xs4[D_ + idx.y];
            const v4f_t g2  = xs4[idx.z];
            const v4f_t g2h = xs4[D_ + idx.z];
            const v4f_t g3  = xs4[idx.w];
            const v4f_t g3h = xs4[D_ + idx.w];

            // col0 += x0*w0.x + x1*w0.z ; col1 += x0*w0.y + x1*w0.w
            // col2 += x2*w1.x + x3*w1.z ; col3 += x2*w1.y + x3*w1.w
#pragma unroll
            for (int g = 0; g < 4; ++g) {
                acc[0][g] = pk_fma_b0(GPART(g0, g0h, g), w0L,
                            pk_fma_b0(GPART(g1, g1h, g), w0H, acc[0][g]));
                acc[1][g] = pk_fma_b1(GPART(g0, g0h, g), w0L,
                            pk_fma_b1(GPART(g1, g1h, g), w0H, acc[1][g]));
                acc[2][g] = pk_fma_b0(GPART(g2, g2h, g), w1L,
                            pk_fma_b0(GPART(g3, g3h, g), w1H, acc[2][g]));
                acc[3][g] = pk_fma_b1(GPART(g2, g2h, g), w1L,
                            pk_fma_b1(GPART(g3, g3h, g), w1H, acc[3][g]));
            }
        }

        const v4f_t bs = *(const v4f_t*)(bsum + col0);
#pragma unroll
        for (int r = 0; r < ROWS; ++r) {
            const int g = r >> 1, h = r & 1;
            v4f_t o = (v4f_t){acc[0][g][h], acc[1][g][h], acc[2][g][h], acc[3][g][h]} + bs;
            // Non-temporal: y is a 128 MB one-shot stream; keep tables hot in L2.
            __builtin_nontemporal_store(o, (v4f_t*)(y + (size_t)(row0 + r) * D_ + col0));
        }
    }
}

// bsum[d] = sum_a b[a, d]   (b is (A, D), batch-independent: reduce once)
__global__ __launch_bounds__(TPB) void bias_sum_kernel(const float* __restrict__ b,
                                                       float* __restrict__ bsum) {
    int d = blockIdx.x * TPB + threadIdx.x;
    if (d < D_) {
        float s = 0.0f;
#pragma unroll
        for (int a = 0; a < A_; ++a) s += b[(size_t)a * D_ + d];
        bsum[d] = s;
    }
}

extern "C" void kernel_launch(void* const* d_in, const int* in_sizes, int n_in,
                              void* d_out, int out_size, void* d_ws, size_t ws_size,
                              hipStream_t stream) {
    const float* x     = (const float*)d_in[0];   // (B, D)
    const float* W     = (const float*)d_in[1];   // (A, NB, 2, 2)
    const float* b     = (const float*)d_in[2];   // (A, D)
    const int*   perms = (const int*)  d_in[3];   // (A, D)
    float*       y     = (float*)d_out;           // (B, D)
    float*       bsum  = (float*)d_ws;            // D floats = 16 KB scratch

    bias_sum_kernel<<<D_ / TPB, TPB, 0, stream>>>(b, bsum);
    pairlin_kernel<<<B_ / ROWS, TPB, 0, stream>>>(x, W, perms, bsum, y);
}